// DecodePredictions_1047972020272
// MI455X (gfx1250) — compile-verified
//
#include <hip/hip_runtime.h>
#include <hip/hip_bf16.h>
#include <math.h>

#define BATCH       8
#define A_TOTAL     76725
#define NUM_CLASSES 80
#define PRED_STRIDE 84
#define KPC         100
#define MAX_DET     100
#define SCORE_THR   0.05f
#define IOU_THR     0.5f

typedef float    v2f  __attribute__((ext_vector_type(2)));
typedef float    v8f  __attribute__((ext_vector_type(8)));
typedef unsigned u32x4 __attribute__((ext_vector_type(4)));
typedef int      i32x4 __attribute__((ext_vector_type(4)));
typedef int      i32x8 __attribute__((ext_vector_type(8)));

// Order-preserving float -> uint key (larger key == larger float)
__device__ __forceinline__ unsigned keyOf(float f) {
    unsigned u = __float_as_uint(f);
    return (u & 0x80000000u) ? ~u : (u | 0x80000000u);
}

// ---------------------------------------------------------------------------
// TDM: DMA one strided fp32 column (stride 84 elems) of `rows` rows from
// global into contiguous LDS at byte offset ldsOff.
// D# per CDNA5 ISA (group0 128b, group1 256b, 2-D tile).
// ---------------------------------------------------------------------------
__device__ __forceinline__ void tdm_load_column(const float* gbase,
                                                unsigned ldsOff, unsigned rows) {
    unsigned long long ga = (unsigned long long)(uintptr_t)gbase;
    u32x4 g0;
    g0.x = 1u;                                          // count=1, no gather
    g0.y = ldsOff;                                      // lds_addr [63:32]
    g0.z = (unsigned)(ga & 0xFFFFFFFFull);              // global_addr lo
    g0.w = (unsigned)((ga >> 32) & 0x1FFFFFFull)        // global_addr hi (57b)
         | (2u << 30);                                  // type=2 ("image")
    i32x8 g1;
    g1[0] = (int)(2u << 16);                            // data_size=2 (4 bytes)
    g1[1] = (int)((84u & 0xFFFFu) << 16);               // tensor_dim0=84 @bit48
    g1[2] = (int)((rows & 0xFFFFu) << 16);              // dim0 hi=0 | tensor_dim1 lo
    g1[3] = (int)(((rows >> 16) & 0xFFFFu)              // tensor_dim1 hi
         | (1u << 16));                                 // tile_dim0=1
    g1[4] = (int)(rows & 0xFFFFu);                      // tile_dim1=rows, tile_dim2=0
    g1[5] = 84;                                         // tensor_dim0_stride lo
    g1[6] = 0;                                          // dim0_stride hi | dim1_stride lo
    g1[7] = 0;                                          // tensor_dim1_stride hi
    i32x4 z4 = {0, 0, 0, 0};
#if defined(__clang_major__) && (__clang_major__ >= 23)
    i32x8 z8 = {0, 0, 0, 0, 0, 0, 0, 0};
    __builtin_amdgcn_tensor_load_to_lds(g0, g1, z4, z4, z8, 0);
#else
    __builtin_amdgcn_tensor_load_to_lds(g0, g1, z4, z4, 0);
#endif
}

// ---------------------------------------------------------------------------
// Kernel 1: decode anchors + deltas -> corner boxes  [B, A, 4]
// ---------------------------------------------------------------------------
__global__ __launch_bounds__(256) void decode_boxes_kernel(
        const float* __restrict__ pred, float* __restrict__ boxes) {
    int gid = blockIdx.x * blockDim.x + threadIdx.x;
    if (gid >= BATCH * A_TOTAL) return;
    int a = gid % A_TOTAL;

    int off, stride, fw;
    if      (a < 57600) { off = 0;     stride = 8;   fw = 80; }
    else if (a < 72000) { off = 57600; stride = 16;  fw = 40; }
    else if (a < 75600) { off = 72000; stride = 32;  fw = 20; }
    else if (a < 76500) { off = 75600; stride = 64;  fw = 10; }
    else                { off = 76500; stride = 128; fw = 5;  }

    int idx  = a - off;
    int d    = idx % 9;
    int cell = idx / 9;
    int gx   = cell % fw, gy = cell / fw;
    float cx = (gx + 0.5f) * (float)stride;
    float cy = (gy + 0.5f) * (float)stride;

    float side = 4.0f * (float)stride;
    const float sqrt_ratio[3] = {0.70710678f, 1.0f, 1.41421356f};
    int ri = d / 3, si = d % 3;
    float sclf = exp2f((float)si * (1.0f / 3.0f));
    float aw = side * sqrt_ratio[ri];
    float ah = side / sqrt_ratio[ri];
    float w  = sclf * aw, h = sclf * ah;

    const float* p = pred + (long long)gid * PRED_STRIDE;
    __builtin_prefetch(p + 256 * PRED_STRIDE, 0, 0);   // global_prefetch_b8
    float dx = p[0] * 0.1f, dy = p[1] * 0.1f;
    float dw = p[2] * 0.2f, dh = p[3] * 0.2f;
    float bx = dx * w + cx, by = dy * h + cy;
    float bw = expf(dw) * w, bh = expf(dh) * h;

    float* o = boxes + (long long)gid * 4;
    o[0] = bx - bw * 0.5f;
    o[1] = by - bh * 0.5f;
    o[2] = bx + bw * 0.5f;
    o[3] = by + bh * 0.5f;
}

// ---------------------------------------------------------------------------
// exact top-100 radix select (4x 8-bit rounds + collect) over n elements
// ---------------------------------------------------------------------------
__device__ void radix_select_100(const float* __restrict__ base, long long stride, int n,
                                 unsigned* hist, unsigned* sel,   // sel[0]=prefix, sel[1]=remain
                                 float* sc, int* id, int* cnt, int tid) {
    if (tid == 0) { sel[0] = 0u; sel[1] = KPC; }
    for (int p = 0; p < 4; ++p) {
        for (int i = tid; i < 256; i += 256) hist[i] = 0u;
        __syncthreads();
        unsigned prefix = sel[0];
        int shift = 24 - 8 * p;
        int hs = shift + 8;
        for (int a = tid; a < n; a += 256) {
            unsigned k = keyOf(base[(long long)a * stride]);
            bool q = (p == 0) || ((hs < 32) && ((k >> hs) == (prefix >> hs)));
            if (q) atomicAdd(&hist[(k >> shift) & 255u], 1u);
        }
        __syncthreads();
        if (tid == 0) {
            unsigned cum = 0, rem = sel[1];
            int bin = 0;
            for (int i = 255; i >= 0; --i) {
                if (cum + hist[i] >= rem) { bin = i; sel[1] = rem - cum; break; }
                cum += hist[i];
            }
            sel[0] = prefix | ((unsigned)bin << shift);
        }
        __syncthreads();
    }
    if (tid == 0) { cnt[0] = 0; cnt[1] = 0; }
    for (int i = tid; i < 128; i += 256) { sc[i] = -3.402823466e38f; id[i] = 0; }
    __syncthreads();
    unsigned pivot = sel[0];
    int nG = KPC - (int)sel[1];
    int nE = (int)sel[1];
    for (int a = tid; a < n; a += 256) {
        float v = base[(long long)a * stride];
        unsigned k = keyOf(v);
        if (k > pivot) {
            int pos = atomicAdd(&cnt[0], 1);
            if (pos < nG) { sc[pos] = v; id[pos] = a; }
        } else if (k == pivot) {
            int e = atomicAdd(&cnt[1], 1);
            if (e < nE) { sc[nG + e] = v; id[nG + e] = a; }
        }
    }
    __syncthreads();
}

__device__ void bitonic_sort128_desc(float* sc, int* id, int tid) {
    for (int k2 = 2; k2 <= 128; k2 <<= 1) {
        for (int j = k2 >> 1; j > 0; j >>= 1) {
            __syncthreads();
            if (tid < 128) {
                int ixj = tid ^ j;
                if (ixj > tid) {
                    bool dirDesc = ((tid & k2) == 0);
                    float a0 = sc[tid], a1 = sc[ixj];
                    bool sw = dirDesc ? (a0 < a1) : (a0 > a1);
                    if (sw) {
                        sc[tid] = a1; sc[ixj] = a0;
                        int t = id[tid]; id[tid] = id[ixj]; id[ixj] = t;
                    }
                }
            }
        }
    }
    __syncthreads();
}

// ---------------------------------------------------------------------------
// Kernel 2: one block per (batch, class). TDM pulls the strided class column
// into LDS once; 5 selection passes run LDS-resident. WMMA builds the
// pairwise union base (area_i + area_j), aliased over the dead column buffer.
// ---------------------------------------------------------------------------
__global__ __launch_bounds__(256) void nms_per_class_kernel(
        const float* __restrict__ pred, const float* __restrict__ boxes,
        float* __restrict__ cand_boxes, float* __restrict__ cand_scores) {
    extern __shared__ float dynbuf[];     // 76800 floats: logits, later uni
    __shared__ unsigned hist[256];
    __shared__ unsigned sel[2];
    __shared__ int      cnt[2];
    __shared__ float    sc[128];
    __shared__ int      id[128];
    __shared__ float    kb[128][4];
    __shared__ float    area[128];
    __shared__ int      keep[128];

    int tid = threadIdx.x;
    int bc  = blockIdx.x;
    int b   = bc / NUM_CLASSES;
    int c   = bc % NUM_CLASSES;
    const float* gcol = pred + (long long)b * A_TOTAL * PRED_STRIDE + 4 + c;

    float* logits = dynbuf;
    float* uni    = dynbuf;               // reused after selection passes

    // --- TDM: strided column -> contiguous LDS (2 chunks, tile_dim1 <= 65535)
    const unsigned CH0 = 65535u;
    const unsigned CH1 = (unsigned)A_TOTAL - CH0;       // 11190
    if (tid < 32) {                                     // wave 0 issues DMA
        unsigned ldsBase = (unsigned)(uintptr_t)logits;
        tdm_load_column(gcol, ldsBase, CH0);
        tdm_load_column(gcol + (long long)CH0 * PRED_STRIDE,
                        ldsBase + CH0 * 4u, CH1);
        __builtin_amdgcn_s_wait_tensorcnt(0);
    }
    __syncthreads();

    // --- exact top-100 logits (sigmoid is monotone), all passes in LDS
    radix_select_100(logits, 1, A_TOTAL, hist, sel, sc, id, cnt, tid);
    bitonic_sort128_desc(sc, id, tid);

    // --- fetch boxes, sigmoid scores, areas, initial keep
    if (tid < 128) {
        if (tid < KPC) {
            const float* bb = boxes + ((long long)b * A_TOTAL + id[tid]) * 4;
            float x1 = bb[0], y1 = bb[1], x2 = bb[2], y2 = bb[3];
            kb[tid][0] = x1; kb[tid][1] = y1; kb[tid][2] = x2; kb[tid][3] = y2;
            area[tid]  = (x2 - x1) * (y2 - y1);
            float s    = 1.0f / (1.0f + expf(-sc[tid]));
            sc[tid]    = s;
            keep[tid]  = (s > SCORE_THR) ? 1 : 0;
        } else {
            kb[tid][0] = kb[tid][1] = kb[tid][2] = kb[tid][3] = 0.0f;
            area[tid] = 0.0f;
            keep[tid] = 0;
        }
    }
    __syncthreads();   // logits dead beyond this point -> uni may overwrite

    // --- union base uni[i][j] = area[i] + area[j] via v_wmma_f32_16x16x4_f32
    {
        int wave = tid >> 5, lane = tid & 31;
        for (int t = wave; t < 49; t += 8) {
            int r0 = (t / 7) * 16, c0 = (t % 7) * 16;
            v2f aOp, bOp;
            aOp.x = (lane < 16) ? area[r0 + lane] : 0.0f;   // A: K=0 col / pad
            aOp.y = (lane < 16) ? 1.0f : 0.0f;              //    K=1 ones / pad
            bOp.x = (lane < 16) ? 1.0f : area[c0 + (lane - 16)]; // B: K=0,1
            bOp.y = 0.0f;                                        //    K=2,3 pad
            v8f acc = {};
            acc = __builtin_amdgcn_wmma_f32_16x16x4_f32(
                false, aOp, false, bOp, (short)0, acc, false, false);
            for (int kk = 0; kk < 8; ++kk) {
                int row = r0 + ((lane < 16) ? kk : kk + 8);
                int col = c0 + (lane & 15);
                uni[row * 112 + col] = acc[kk];
            }
        }
    }
    __syncthreads();

    // --- greedy NMS: 100 sequential rounds, columns parallel across threads
    for (int i = 0; i < KPC; ++i) {
        __syncthreads();
        if (keep[i]) {
            float x1 = kb[i][0], y1 = kb[i][1], x2 = kb[i][2], y2 = kb[i][3];
            for (int j = tid; j < KPC; j += 256) {
                if (j > i && keep[j]) {
                    float ix1 = fmaxf(x1, kb[j][0]);
                    float iy1 = fmaxf(y1, kb[j][1]);
                    float ix2 = fminf(x2, kb[j][2]);
                    float iy2 = fminf(y2, kb[j][3]);
                    float iw = fmaxf(ix2 - ix1, 0.0f);
                    float ih = fmaxf(iy2 - iy1, 0.0f);
                    float inter = iw * ih;
                    float u = uni[i * 112 + j] - inter;
                    float iou = inter / fmaxf(u, 1e-8f);
                    if (iou > IOU_THR) keep[j] = 0;
                }
            }
        }
    }
    __syncthreads();

    for (int i = tid; i < KPC; i += 256) {
        long long o = (long long)bc * KPC + i;
        cand_scores[o] = keep[i] ? sc[i] : -1.0f;
        float* ob = cand_boxes + o * 4;
        ob[0] = kb[i][0]; ob[1] = kb[i][1]; ob[2] = kb[i][2]; ob[3] = kb[i][3];
    }
}

// ---------------------------------------------------------------------------
// Kernel 3: per batch, merge 80*100 candidates -> global top-100 + outputs
// ---------------------------------------------------------------------------
__global__ __launch_bounds__(256) void merge_topk_kernel(
        const float* __restrict__ cand_scores, const float* __restrict__ cand_boxes,
        float* __restrict__ out_boxes, float* __restrict__ out_scores,
        float* __restrict__ out_classes, int* __restrict__ out_valid) {
    __shared__ unsigned hist[256];
    __shared__ unsigned sel[2];
    __shared__ int      cnt[2];
    __shared__ float    sc[128];
    __shared__ int      id[128];
    __shared__ int      validCnt;

    const int N = NUM_CLASSES * KPC;
    int tid = threadIdx.x;
    int b   = blockIdx.x;
    const float* s = cand_scores + (long long)b * N;

    radix_select_100(s, 1, N, hist, sel, sc, id, cnt, tid);
    bitonic_sort128_desc(sc, id, tid);

    if (tid == 0) validCnt = 0;
    __syncthreads();

    if (tid < MAX_DET) {
        float v  = sc[tid];
        int   ix = id[tid];
        bool valid = v > SCORE_THR;
        if (valid) atomicAdd(&validCnt, 1);
        const float* bb = cand_boxes + ((long long)b * N + ix) * 4;
        float* ob = out_boxes + ((long long)b * MAX_DET + tid) * 4;
        ob[0] = valid ? bb[0] : 0.0f;
        ob[1] = valid ? bb[1] : 0.0f;
        ob[2] = valid ? bb[2] : 0.0f;
        ob[3] = valid ? bb[3] : 0.0f;
        out_scores[b * MAX_DET + tid]  = valid ? v : 0.0f;
        out_classes[b * MAX_DET + tid] = valid ? (float)(ix / KPC) : 0.0f;
    }
    __syncthreads();
    if (tid == 0) out_valid[b] = validCnt;
}

// ---------------------------------------------------------------------------
extern "C" void kernel_launch(void* const* d_in, const int* in_sizes, int n_in,
                              void* d_out, int out_size, void* d_ws, size_t ws_size,
                              hipStream_t stream) {
    const float* pred = (const float*)d_in[1];   // [8, 76725, 84] fp32

    float* wsf         = (float*)d_ws;
    float* boxes       = wsf;                                               // B*A*4
    float* cand_boxes  = boxes + (long long)BATCH * A_TOTAL * 4;            // B*C*K*4
    float* cand_scores = cand_boxes + (long long)BATCH * NUM_CLASSES * KPC * 4;

    float* out         = (float*)d_out;
    float* out_boxes   = out;
    float* out_scores  = out_boxes + BATCH * MAX_DET * 4;
    float* out_classes = out_scores + BATCH * MAX_DET;
    int*   out_valid   = (int*)(out_classes + BATCH * MAX_DET);

    int total = BATCH * A_TOTAL;
    decode_boxes_kernel<<<(total + 255) / 256, 256, 0, stream>>>(pred, boxes);
    // dynamic LDS: 76800 floats = 307200 B (column buffer, later WMMA uni)
    nms_per_class_kernel<<<BATCH * NUM_CLASSES, 256, 307200, stream>>>(
        pred, boxes, cand_boxes, cand_scores);
    merge_topk_kernel<<<BATCH, 256, 0, stream>>>(
        cand_scores, cand_boxes, out_boxes, out_scores, out_classes, out_valid);
}